// HashGridEncoder_43422119362767
// MI455X (gfx1250) — compile-verified
//
#include <hip/hip_runtime.h>

// ---------------------------------------------------------------------------
// HashGridEncoder for MI455X (gfx1250, wave32)
//
// Memory-bound gather kernel: 64 MB of hash tables (fully resident in the
// 192 MB L2), 134 MB write-once output stream -> ~6 us HBM roofline.
// Strategy:
//   * 1 thread per point, 16 levels fully unrolled -> ~64 in-flight b64
//     gathers per lane (deep LOADcnt, hides L2 latency)
//   * hash computed in u32 (mod 2^19 == low 19 bits)
//   * results staged in LDS, then written back fully coalesced (512B / wave
//     instruction) via CDNA5 global_store_async_from_lds_b128 with
//     th:TH_STORE_NT so the output stream does not evict the L2 tables
// ---------------------------------------------------------------------------

#define NUM_LEVELS   16
#define FEAT         2
#define HASH_SIZE    (1u << 19)
#define HASH_MASK    (HASH_SIZE - 1u)
#define PRIME2       2654435761u
#define TPB          256
#define PT_STRIDE    36   // floats per point in LDS: 32 payload + 4 pad (144B row, 16B aligned)

#ifndef USE_ASYNC_STORE
#define USE_ASYNC_STORE 1   // CDNA5 LDS->global async store path
#endif

typedef float v2f __attribute__((ext_vector_type(2)));

__global__ __launch_bounds__(TPB)
void hashgrid_enc_kernel(const float* __restrict__ x,
                         const float* __restrict__ lat,
                         float* __restrict__ out)
{
    __shared__ float smem[TPB * PT_STRIDE];   // 36 KB staging tile

    const unsigned t = threadIdx.x;
    const unsigned n = blockIdx.x * TPB + t;

    // resolutions = int(16 * 1.5^i), exact in f32
    const float res[NUM_LEVELS] = {
        16.f, 24.f, 36.f, 54.f, 81.f, 121.f, 182.f, 273.f,
        410.f, 615.f, 922.f, 1383.f, 2075.f, 3113.f, 4670.f, 7006.f
    };

    // coalesced non-temporal b64 load of this thread's point (read-once stream)
    const v2f p = __builtin_nontemporal_load((const v2f*)x + n);
    const float x0 = (p.x + 1.0f) * 0.5f;   // [-1,1] -> [0,1]
    const float x1 = (p.y + 1.0f) * 0.5f;

#pragma unroll
    for (int l = 0; l < NUM_LEVELS; ++l) {
        const float px = x0 * res[l];
        const float py = x1 * res[l];
        const float fx = floorf(px);
        const float fy = floorf(py);
        const float wx = px - fx;
        const float wy = py - fy;
        const unsigned ix = (unsigned)(int)fx;     // x in [-1,1] => coords >= 0
        const unsigned iy = (unsigned)(int)fy;

        // hash = (cx*1 + cy*PRIME2) mod 2^19 : low 19 bits, exact in u32
        const unsigned hy0 = iy * PRIME2;
        const unsigned hy1 = hy0 + PRIME2;
        const unsigned h00 = (ix      + hy0) & HASH_MASK;   // offset (0,0)
        const unsigned h01 = (ix      + hy1) & HASH_MASK;   // offset (0,1)
        const unsigned h10 = (ix + 1u + hy0) & HASH_MASK;   // offset (1,0)
        const unsigned h11 = (ix + 1u + hy1) & HASH_MASK;   // offset (1,1)

        const float2* __restrict__ tbl = (const float2*)lat + (size_t)l * HASH_SIZE;
        const float2 f00 = tbl[h00];   // 4x global_load_b64, L2-resident tables
        const float2 f01 = tbl[h01];
        const float2 f10 = tbl[h10];
        const float2 f11 = tbl[h11];

        const float w00 = (1.0f - wx) * (1.0f - wy);
        const float w01 = (1.0f - wx) * wy;
        const float w10 = wx * (1.0f - wy);
        const float w11 = wx * wy;

        const float o0 = f00.x * w00 + f01.x * w01 + f10.x * w10 + f11.x * w11;
        const float o1 = f00.y * w00 + f01.y * w01 + f10.y * w10 + f11.y * w11;

        // stage per-level float2 into LDS (8B aligned: t*144 + l*8)
        *(float2*)&smem[t * PT_STRIDE + 2 * l] = make_float2(o0, o1);
    }

    __syncthreads();   // s_wait_dscnt + barrier: LDS tile complete

    // Coalesced writeback: iteration j, lanes of a wave cover 512B contiguous.
    // thread t = 8a + b handles local point (j*32 + a), feature quad b.
    const unsigned a = t >> 3;
    const unsigned b = t & 7;
#pragma unroll
    for (int j = 0; j < 8; ++j) {
        const unsigned pl  = (unsigned)j * 32u + a;                    // local point
        const unsigned lfo = pl * PT_STRIDE + b * 4u;                  // LDS float offset
        const unsigned gfo = (blockIdx.x * TPB + pl) * 32u + b * 4u;   // global float offset
#if USE_ASYNC_STORE
        // Generic LDS pointer is {aperture_hi32, lds_offset32}; flat->LDS
        // truncates to addr[31:0] (ISA 10.2), so low 32 bits = LDS byte addr.
        // Taking the address also marks smem as escaped -> staging stores live.
        const unsigned lds_byte = (unsigned)(uintptr_t)&smem[lfo];   // 16B aligned
        const unsigned g_byte   = gfo * 4u;          // 16B aligned; out < 4GB
        // CDNA5 async LDS->global store, GVS mode: VADDR=u32 offset, VSRC=LDS
        // byte addr, SADDR=64-bit uniform base. Tracked by ASYNCcnt.
        // th:TH_STORE_NT: write-once stream, keep hash tables resident in L2.
        asm volatile("global_store_async_from_lds_b128 %0, %1, %2 th:TH_STORE_NT"
                     :: "v"(g_byte), "v"(lds_byte), "s"(out)
                     : "memory");
#else
        const float4 v = *(const float4*)&smem[lfo];
        __builtin_nontemporal_store(v, (float4*)(out + gfo));
#endif
    }
#if USE_ASYNC_STORE
    asm volatile("s_wait_asynccnt 0" ::: "memory");
#endif
}

extern "C" void kernel_launch(void* const* d_in, const int* in_sizes, int n_in,
                              void* d_out, int out_size, void* d_ws, size_t ws_size,
                              hipStream_t stream) {
    const float* x   = (const float*)d_in[0];   // [N,2] f32
    const float* lat = (const float*)d_in[1];   // [16*2^19, 2] f32
    float* out = (float*)d_out;                 // [N, 32] f32

    const int N = in_sizes[0] / 2;              // 2^20
    const int blocks = N / TPB;                 // 4096, N is a multiple of 256

    hipLaunchKernelGGL(hashgrid_enc_kernel, dim3(blocks), dim3(TPB), 0, stream,
                       x, lat, out);
}